// MyLayer_6073083757041
// MI455X (gfx1250) — compile-verified
//
#include <hip/hip_runtime.h>
#include <stdint.h>

// out[i] = w[i*16 + 15] * image[i*3]  — streaming elementwise multiply.
// Bandwidth-bound (~1.28 GB min HBM traffic -> ~55us at 23.3 TB/s).
// CDNA5 path: TDM gathers the stride-16 w stream into dense LDS tiles
// (double-buffered per wave, s_wait_tensorcnt-paced); image is read with
// aligned non-temporal b128 loads (3 x float4 cover 4 outputs' channel-0
// samples); output uses non-temporal b128 stores. NT policy keeps the
// 1.2 GB stream-once traffic from thrashing the 192 MB L2.

typedef __attribute__((ext_vector_type(4))) unsigned int tdm_v4u;
typedef __attribute__((ext_vector_type(8))) int tdm_v8i;
typedef __attribute__((ext_vector_type(4))) int tdm_v4i;
typedef __attribute__((ext_vector_type(4))) float f32x4;

#define TILE 512                 // outputs staged per wave-tile (2 KB LDS)
#define WAVES_PER_BLOCK 8
#define BLOCK_THREADS (WAVES_PER_BLOCK * 32)
#define GRID_BLOCKS 1024

// TDM gather: `rows` elements, one float every `stride_elems` floats,
// compacted into LDS at lds_addr. D# per CDNA5 ISA.
__device__ __forceinline__ void tdm_gather_stride(uint64_t gaddr_bytes,
                                                  unsigned lds_addr,
                                                  int rows,
                                                  unsigned stride_elems) {
  tdm_v4u g0;
  g0.x = 1u;                                        // count=1 (valid), user mode
  g0.y = lds_addr;                                  // lds_addr [63:32]
  g0.z = (unsigned)(gaddr_bytes & 0xffffffffull);   // global_addr[31:0]
  g0.w = ((unsigned)((gaddr_bytes >> 32) & 0x01ffffffull)) | (2u << 30); // addr[56:32] | type=2

  tdm_v8i g1;
  g1[0] = (int)(2u << 16);                   // data_size=2 (4B); no mask/pad/iterate
  g1[1] = (int)(stride_elems << 16);         // tensor_dim0 = stride (x=0 in range)
  g1[2] = (int)(0xffffu << 16);              // tensor_dim1 lo16 (huge -> no OOB)
  g1[3] = (int)(0x7fffu | (1u << 16));       // tensor_dim1 hi16 | tile_dim0 = 1
  g1[4] = rows;                              // tile_dim1 = rows, tile_dim2 = 0
  g1[5] = (int)stride_elems;                 // tensor_dim0_stride lo32
  g1[6] = 0;
  g1[7] = 0;

  tdm_v4i g2 = {0, 0, 0, 0};                 // dims 2/3 unused
  tdm_v4i g3 = {0, 0, 0, 0};
#if defined(__clang_major__) && __clang_major__ >= 23
  tdm_v8i g4 = {0, 0, 0, 0, 0, 0, 0, 0};
  __builtin_amdgcn_tensor_load_to_lds(g0, g1, g2, g3, g4, 0);
#else
  __builtin_amdgcn_tensor_load_to_lds(g0, g1, g2, g3, 0);
#endif
}

__global__ __launch_bounds__(BLOCK_THREADS)
void slice_mul_kernel(const float* __restrict__ img, const float* __restrict__ w,
                      float* __restrict__ out, int n, int tilesPerWave) {
  __shared__ float wstage[WAVES_PER_BLOCK][2][TILE];

  const int lane = (int)(threadIdx.x & 31u);
  const int wave = __builtin_amdgcn_readfirstlane((int)(threadIdx.x >> 5));
  const int waveId = (int)blockIdx.x * WAVES_PER_BLOCK + wave;
  const int mainTiles = tilesPerWave * (int)gridDim.x * WAVES_PER_BLOCK;

  const uint64_t wbytes = (uint64_t)(uintptr_t)w;
  const unsigned ldsbase = (unsigned)(uintptr_t)(&wstage[wave][0][0]);

  if (tilesPerWave > 0) {
    const int t0 = waveId * tilesPerWave;
    // prologue: stage tile t0 into buffer 0
    tdm_gather_stride(wbytes + ((uint64_t)t0 * TILE * 16u + 15u) * 4u,
                      ldsbase, TILE, 16u);
    for (int k = 0; k < tilesPerWave; ++k) {
      const int t = t0 + k;
      const int cur = k & 1;
      if (k + 1 < tilesPerWave) {
        // async prefetch of next tile's w gather into the other buffer
        tdm_gather_stride(wbytes + ((uint64_t)(t + 1) * TILE * 16u + 15u) * 4u,
                          ldsbase + (unsigned)(((k + 1) & 1) * TILE * 4), TILE, 16u);
        // prefetch next tile's image span (512*12 B) — one line per lane
        __builtin_prefetch(&img[(size_t)(t + 1) * TILE * 3 + (size_t)lane * 48], 0, 1);
        __builtin_amdgcn_s_wait_tensorcnt((short)1);   // current tile's DMA done
      } else {
        __builtin_amdgcn_s_wait_tensorcnt((short)0);   // drain
      }
      const int base = t * TILE;
      const float* wbuf = &wstage[wave][cur][0];
#pragma unroll
      for (int q = 0; q < TILE / 128; ++q) {           // 4 outputs per lane per q
        const int j0 = q * 128 + lane * 4;
        const size_t i0 = (size_t)base + (size_t)j0;   // 16B-aligned output index
        // image: 4 channel-0 samples live at floats {0,3,6,9} of a 48B block
        const f32x4* ip = (const f32x4*)&img[i0 * 3];  // 48B-aligned
        const f32x4 a = __builtin_nontemporal_load(ip + 0);
        const f32x4 b = __builtin_nontemporal_load(ip + 1);
        const f32x4 c = __builtin_nontemporal_load(ip + 2);
        const f32x4 wv = *(const f32x4*)&wbuf[j0];     // dense ds_load_b128
        f32x4 r;
        r.x = a.x * wv.x;   // img[(i0+0)*3]
        r.y = a.w * wv.y;   // img[(i0+1)*3] = float 3
        r.z = b.z * wv.z;   // img[(i0+2)*3] = float 6
        r.w = c.y * wv.w;   // img[(i0+3)*3] = float 9
        __builtin_nontemporal_store(r, (f32x4*)&out[i0]);
      }
    }
  }

  // Generic tail (empty for n = 4096*4096 with this launch config)
  for (size_t i = (size_t)mainTiles * TILE + (size_t)blockIdx.x * BLOCK_THREADS + threadIdx.x;
       i < (size_t)n; i += (size_t)gridDim.x * BLOCK_THREADS) {
    out[i] = img[i * 3] * w[i * 16 + 15];
  }
}

extern "C" void kernel_launch(void* const* d_in, const int* in_sizes, int n_in,
                              void* d_out, int out_size, void* d_ws, size_t ws_size,
                              hipStream_t stream) {
  (void)in_sizes; (void)n_in; (void)d_ws; (void)ws_size;
  const float* img = (const float*)d_in[0];   // (4096,4096,3) f32
  const float* w   = (const float*)d_in[1];   // (4096,4096,16) f32
  float* out = (float*)d_out;                 // (4096,4096) f32
  const int n = out_size;                     // 16,777,216
  const int wavesTotal = GRID_BLOCKS * WAVES_PER_BLOCK;       // 8192
  const int tilesTotal = n / TILE;                            // 32768
  const int tilesPerWave = tilesTotal / wavesTotal;           // 4 (host-side div)
  slice_mul_kernel<<<GRID_BLOCKS, BLOCK_THREADS, 0, stream>>>(img, w, out, n,
                                                              tilesPerWave);
}